// Spline5DInterpolant_35476429865445
// MI455X (gfx1250) — compile-verified
//
#include <hip/hip_runtime.h>

// CDNA5 / gfx1250: wave32, WMMA 16x16 shapes.
typedef float v2f __attribute__((ext_vector_type(2)));
typedef float v8f __attribute__((ext_vector_type(8)));

#define NGRID 32
#define S4 32          // stride of dim 3 (0-based) in elements
#define S3 1024        // 32^2
#define S2 32768       // 32^3
#define S1 1048576     // 32^4

__device__ __forceinline__ float pick4(const float w[4], int i) {
  // small select chain (i is lane-varying but in 0..3)
  float r = w[0];
  r = (i == 1) ? w[1] : r;
  r = (i == 2) ? w[2] : r;
  r = (i == 3) ? w[3] : r;
  return r;
}

__global__ __launch_bounds__(32) void spline5d_wmma_kernel(
    const float* __restrict__ xq,    // [B][5]
    const float* __restrict__ cf,    // [32^5]
    const float* __restrict__ alo,   // [5]
    const float* __restrict__ bhi,   // [5]
    const float* __restrict__ nintv, // [5]
    float* __restrict__ out,         // [B]
    int npts)
{
  const int p = blockIdx.x;
  if (p >= npts) return;                 // wave-uniform branch; EXEC all-1s below

  const int lane = threadIdx.x & 31;
  const int half = lane >> 4;            // 0: A-cols {0,1}/B-rows {0,1}; 1: {2,3}
  const int m    = lane & 15;            // A-matrix row (16x4 tile)
  const int gl   = m >> 2;               // local j123-block within tile == i3
  const int j4   = m & 3;

  // ---- per-dim base index + the 4 non-zero cubic B-spline weights ----
  // t_j = |frac + 1 - j|:  w0=(1-f)^3, w1=4-6f^2+3f^3, w2=inner(1-f), w3=f^3
  int   o[5];
  float w[5][4];
  #pragma unroll
  for (int k = 0; k < 5; ++k) {
    float h  = (bhi[k] - alo[k]) / nintv[k];
    float u  = (xq[p * 5 + k] - alo[k]) / h;
    int   oi = (int)floorf(u);
    oi = oi < 0 ? 0 : oi;
    oi = oi > (NGRID - 4) ? (NGRID - 4) : oi;
    float f = u - (float)oi;
    float g = 1.0f - f;
    w[k][0] = g * g * g;
    w[k][1] = 4.0f - 6.0f * f * f + 3.0f * f * f * f;
    w[k][2] = 4.0f - 6.0f * g * g + 3.0f * g * g * g;
    w[k][3] = f * f * f;
    o[k] = oi;
  }

  // lane-fixed part of the address and of the weight product
  const float* cbase = cf
      + (size_t)o[0] * S1 + (size_t)o[1] * S2 + (size_t)o[2] * S3
      + (size_t)o[3] * S4 + (size_t)o[4]
      + (size_t)(gl * S3 + j4 * S4 + 2 * half);
  const float wlane = pick4(w[2], gl) * pick4(w[3], j4);   // w3(i3)*w4(j4)

  // ---- B matrix (4x16): every column carries the w5 weights ----
  v2f bmat;
  bmat.x = half ? w[4][2] : w[4][0];
  bmat.y = half ? w[4][3] : w[4][1];

  // ---- gather + prescale A tiles (loads all independent -> issue early) ----
  // G = 4*it + gl  =>  i3 = gl (folded above), i2 = it&3, i1 = it>>2 (constant
  // indices after unroll: weight lookups stay in registers).
  v2f av[16];
  #pragma unroll
  for (int it = 0; it < 16; ++it) {
    const int i1 = it >> 2;
    const int i2 = it & 3;
    const float* r = cbase + i1 * S1 + i2 * S2;
    const float sc = w[0][i1] * w[1][i2] * wlane;          // w1*w2*w3*w4
    av[it].x = r[0] * sc;
    av[it].y = r[1] * sc;
  }

  // ---- 16 accumulating fp32 WMMAs: D[m,n] += sum_k A[m,k]*w5[k] ----
  v8f acc = {};
  #pragma unroll
  for (int it = 0; it < 16; ++it) {
    acc = __builtin_amdgcn_wmma_f32_16x16x4_f32(
        /*neg_a=*/false, av[it], /*neg_b=*/false, bmat,
        /*c_mod=*/(short)0, acc, /*reuse_a=*/false, /*reuse_b=*/false);
  }

  // ---- reduce over the 16 M-rows of one column ----
  // lane L<16 holds D[0..7, L]; lane L+16 holds D[8..15, L]
  float partial = acc[0] + acc[1] + acc[2] + acc[3]
                + acc[4] + acc[5] + acc[6] + acc[7];
  partial += __shfl_xor(partial, 16, 32);
  if (lane == 0) out[p] = partial;
}

extern "C" void kernel_launch(void* const* d_in, const int* in_sizes, int n_in,
                              void* d_out, int out_size, void* d_ws, size_t ws_size,
                              hipStream_t stream) {
  const float* x = (const float*)d_in[0];   // [B,5]
  const float* c = (const float*)d_in[1];   // [32^5]
  const float* a = (const float*)d_in[2];   // [5]
  const float* b = (const float*)d_in[3];   // [5]
  const float* n = (const float*)d_in[4];   // [5]
  float* out = (float*)d_out;
  const int npts = in_sizes[0] / 5;         // B = 64
  spline5d_wmma_kernel<<<npts, 32, 0, stream>>>(x, c, a, b, n, out, npts);
}